// DeformConv_90735479095316
// MI455X (gfx1250) — compile-verified
//
#include <hip/hip_runtime.h>
#include <stdint.h>

#define KK 3
#define TILE 256            // pixels per workgroup (one row segment)
#define WAVE 32             // wave32
#define NCH 18              // 2*K*K offset channels

typedef __attribute__((ext_vector_type(4))) unsigned int u32x4;
typedef __attribute__((ext_vector_type(8))) unsigned int u32x8;

__device__ __forceinline__ float bilin(const float* __restrict__ img,
                                       float y, float x, int H, int W) {
    float y0f = floorf(y), x0f = floorf(x);
    float ly = y - y0f,  lx = x - x0f;
    float hy = 1.0f - ly, hx = 1.0f - lx;
    int y0 = (int)y0f, x0 = (int)x0f;
    int y1 = y0 + 1,   x1 = x0 + 1;

    auto corner = [&](int yi, int xi) -> float {
        bool v = (yi >= 0) & (yi < H) & (xi >= 0) & (xi < W);
        int yc = yi < 0 ? 0 : (yi > H - 1 ? H - 1 : yi);
        int xc = xi < 0 ? 0 : (xi > W - 1 ? W - 1 : xi);
        float g = img[(size_t)yc * W + xc];
        return v ? g : 0.0f;
    };

    return hy * hx * corner(y0, x0) + hy * lx * corner(y0, x1) +
           ly * hx * corner(y1, x0) + ly * lx * corner(y1, x1);
}

__global__ __launch_bounds__(TILE)
void deform_conv_tdm_kernel(const float* __restrict__ inp,     // (B,H,W)
                            const float* __restrict__ weight,  // (9,)
                            const float* __restrict__ offset,  // (B,18,H,W)
                            float* __restrict__ out,           // (B,H,W)
                            int H, int W) {
    // Per-wave LDS slab: 18 channels x 32 lanes of f32 = 2304 B; 8 waves = 18 KB.
    __shared__ float smem[NCH * TILE];

    const int tile_x = blockIdx.x;
    const int h      = blockIdx.y;
    const int b      = blockIdx.z;
    const int w0     = tile_x * TILE;
    const int lane   = threadIdx.x & (WAVE - 1);
    const int wave   = __builtin_amdgcn_readfirstlane((int)(threadIdx.x >> 5));

    const uint64_t HW = (uint64_t)H * (uint64_t)W;

    // ---- Build TDM descriptor (D#) for this wave's 18x32 f32 offset tile ----
    // Tile start in global memory: offset[b, 0, h, w0 + wave*32]
    const uint64_t gaddr = (uint64_t)(uintptr_t)offset +
        4ull * ((uint64_t)b * (uint64_t)NCH * HW +
                (uint64_t)h * (uint64_t)W +
                (uint64_t)(w0 + wave * WAVE));

    // LDS byte offset of this wave's slab (generic ptr low 32 bits == LDS offset)
    const uint32_t lds_addr =
        (uint32_t)(uintptr_t)(void*)smem + (uint32_t)wave * (NCH * WAVE * 4u);

    const uint32_t td0      = WAVE;              // tensor_dim0 (elements)
    const uint32_t td1      = NCH;               // tensor_dim1 (rows)
    const uint32_t tile0    = WAVE;              // tile_dim0
    const uint32_t tile1    = NCH;               // tile_dim1
    const uint64_t stride0  = HW;                // row stride in elements (channel plane)

    u32x4 g0;
    g0.x = 1u;                                                  // count=1, user mode
    g0.y = lds_addr;                                            // lds_addr[31:0]
    g0.z = (uint32_t)(gaddr & 0xFFFFFFFFull);                   // global_addr[31:0]
    g0.w = (uint32_t)((gaddr >> 32) & 0x01FFFFFFull) | (2u << 30); // ga[56:32] | type=2

    u32x8 g1;
    g1.s0 = 0x00020000u;                                        // data_size=2 (4 bytes)
    g1.s1 = (td0 & 0xFFFFu) << 16;                              // tensor_dim0[15:0]
    g1.s2 = ((td0 >> 16) & 0xFFFFu) | ((td1 & 0xFFFFu) << 16);  // td0[31:16] | td1[15:0]
    g1.s3 = ((td1 >> 16) & 0xFFFFu) | ((tile0 & 0xFFFFu) << 16);// td1[31:16] | tile_dim0
    g1.s4 = (tile1 & 0xFFFFu);                                  // tile_dim1 | tile_dim2=0
    g1.s5 = (uint32_t)(stride0 & 0xFFFFFFFFull);                // dim0_stride[31:0]
    g1.s6 = (uint32_t)((stride0 >> 32) & 0xFFFFull);            // dim0_stride[47:32]
    g1.s7 = 0u;                                                 // dim1_stride (unused, 2D)

    // Issue TDM load, non-temporal: the 151 MB offset stream is touched exactly
    // once; keep WGP$/L2 capacity for the repeatedly-gathered input image.
    // (2-group form: VADDR2/VADDR3 = NULL -> 2D tensor)
    asm volatile("tensor_load_to_lds %0, %1 th:TH_LOAD_NT" :: "s"(g0), "s"(g1) : "memory");

    // ---- While DMA runs, hoist the 9 weights (uniform -> scalar loads) ----
    float wk[KK * KK];
#pragma unroll
    for (int k = 0; k < KK * KK; ++k) wk[k] = weight[k];

    // Wait for this wave's tensor DMA, then consume our private slab (no barrier
    // needed: each wave reads only the slab it loaded).
    __builtin_amdgcn_s_wait_tensorcnt(0);

    const float* slab = &smem[wave * NCH * WAVE];
    const float* img  = inp + (size_t)b * HW;
    const int w       = w0 + wave * WAVE + lane;

    float acc = 0.0f;
#pragma unroll
    for (int k = 0; k < KK * KK; ++k) {
        float oy = slab[(2 * k + 0) * WAVE + lane];
        float ox = slab[(2 * k + 1) * WAVE + lane];
        float y = (float)h + (float)(k / KK - 1) + oy;
        float x = (float)w + (float)(k % KK - 1) + ox;
        acc = fmaf(wk[k], bilin(img, y, x, H, W), acc);
    }

    // Output is written once and never re-read: stream it past the caches.
    __builtin_nontemporal_store(acc, &out[(size_t)b * HW + (size_t)h * W + w]);
}

extern "C" void kernel_launch(void* const* d_in, const int* in_sizes, int n_in,
                              void* d_out, int out_size, void* d_ws, size_t ws_size,
                              hipStream_t stream) {
    const float* inp    = (const float*)d_in[0];   // (B,H,W) f32
    const float* weight = (const float*)d_in[1];   // (1,1,3,3) f32
    const float* offset = (const float*)d_in[2];   // (B,18,H,W) f32
    float* out          = (float*)d_out;

    const int H = 512, W = 512;
    const int B = in_sizes[0] / (H * W);           // 8

    dim3 grid(W / TILE, H, B);                     // (2, 512, 8)
    dim3 block(TILE);                              // 256 threads = 8 wave32
    deform_conv_tdm_kernel<<<grid, block, 0, stream>>>(inp, weight, offset, out, H, W);
}